// GAT_28157805593447
// MI455X (gfx1250) — compile-verified
//
#include <hip/hip_runtime.h>
#include <hip/hip_bf16.h>
#include <math.h>

#define IN_DIM   128
#define HID      16
#define HEADS    8
#define OUT_DIM  40
#define NEG_SLOPE 0.2f

typedef __attribute__((ext_vector_type(2))) float v2f;
typedef __attribute__((ext_vector_type(4))) float v4f;
typedef __attribute__((ext_vector_type(8))) float v8f;

// ---------------------------------------------------------------- utilities
static __device__ __forceinline__ void atomicMaxF(float* addr, float val) {
    // Valid for all sign combinations when *addr initialized to -inf:
    // nonneg floats order like signed ints; neg floats order reversed as unsigned.
    if (val >= 0.0f) {
        atomicMax((int*)addr, __float_as_int(val));
    } else {
        atomicMin((unsigned int*)addr, (unsigned int)__float_as_int(val));
    }
}

__global__ void k_fill(float* __restrict__ p, float v, int n) {
    int i = blockIdx.x * blockDim.x + threadIdx.x;
    if (i < n) p[i] = v;
}

// ------------------------------------------------------- fp32 WMMA GEMM
// C[nrows x ncols] = A[nrows x K] * B[K x ncols], all row-major.
// grid.x = ceil(nrows/16); blockDim.x = 32 * ceil(ncols/16); wave w -> col tile w.
// Uses V_WMMA_F32_16X16X4_F32 (exact f32 accumulate). Inner loop is pure
// pointer-increment addressing, no divergent guards.
__global__ void k_gemm_wmma_f32(const float* __restrict__ A, const float* __restrict__ B,
                                float* __restrict__ C, int nrows, int K, int ncols) {
    const int lane  = threadIdx.x & 31;
    const int wave  = threadIdx.x >> 5;
    const int row0  = blockIdx.x << 4;
    const int m     = lane & 15;
    const int half  = lane >> 4;          // 0 or 1
    const int kh    = half << 1;          // K sub-offset {0,2}
    const int colN  = (wave << 4) + m;
    const bool colOK = (colN < ncols);

    // Out-of-range lanes read column 0 (safe) and zero their B values; a lane's
    // B only ever feeds the output column that same lane holds in D, which is
    // exactly the column we skip at store time.
    const float bmask = colOK ? 1.0f : 0.0f;
    const int   colB  = colOK ? colN : 0;

    int arow = row0 + m;
    if (arow >= nrows) arow = 0;          // safe dummy read; stores are guarded

    const float* Ap = A + (size_t)arow * K + kh;       // advances by 4 per step
    const float* Bp = B + (size_t)kh * ncols + colB;   // advances by 4*ncols
    const size_t bstep = (size_t)4 * ncols;

    v8f acc = {0.f, 0.f, 0.f, 0.f, 0.f, 0.f, 0.f, 0.f};
#pragma unroll 4
    for (int k0 = 0; k0 < K; k0 += 4) {
        v2f a = *(const v2f*)Ap;          // A 16x4: row lane&15, K pair kh..kh+1
        v2f b;                            // B 4x16: col lane&15, same K pair
        b[0] = Bp[0]     * bmask;
        b[1] = Bp[ncols] * bmask;
        Ap += 4;
        Bp += bstep;
        acc = __builtin_amdgcn_wmma_f32_16x16x4_f32(false, a, false, b, (short)0, acc,
                                                    false, false);
    }
    if (colOK) {
        const int rbase = row0 + (half << 3);   // C/D: row = v + 8*(lane>>4)
        for (int v = 0; v < 8; ++v) {
            int r = rbase + v;
            if (r < nrows) C[(size_t)r * ncols + colN] = acc[v];
        }
    }
}

// --------------------------------------------- attention scalar products
// a_src[n,h] = sum_d h[n,h,d]*att_src[h,d]; same for dst. One thread per (n,h).
__global__ void k_att(const float* __restrict__ h, const float* __restrict__ att_src,
                      const float* __restrict__ att_dst, float* __restrict__ a_src,
                      float* __restrict__ a_dst, int Nn, int H, int D) {
    int t = blockIdx.x * blockDim.x + threadIdx.x;
    if (t >= Nn * H) return;
    int n = t / H, hh = t % H;
    const float* hp = h + (size_t)n * H * D + (size_t)hh * D;
    const float* as = att_src + hh * D;
    const float* ad = att_dst + hh * D;
    float ss = 0.f, sd = 0.f;
    for (int d = 0; d < D; ++d) { float v = hp[d]; ss += v * as[d]; sd += v * ad[d]; }
    a_src[t] = ss;
    a_dst[t] = sd;
}

// ------------------------------------------------------- edge softmax passes
static __device__ __forceinline__ void edge_sd(const int* __restrict__ ei, int E, int k,
                                               int* s, int* d) {
    if (k < E) { *s = ei[k]; *d = ei[E + k]; }
    else       { *s = k - E; *d = k - E; }     // self loops appended
}

__global__ void k_edge_max(const int* __restrict__ ei, int E, int ET,
                           const float* __restrict__ a_src, const float* __restrict__ a_dst,
                           float* __restrict__ mmax, int H) {
    int t = blockIdx.x * blockDim.x + threadIdx.x;
    if (t >= ET * H) return;
    int h = t % H, k = t / H;
    int s, d; edge_sd(ei, E, k, &s, &d);
    float e = a_src[s * H + h] + a_dst[d * H + h];
    e = (e > 0.f) ? e : NEG_SLOPE * e;
    atomicMaxF(&mmax[d * H + h], e);
}

__global__ void k_edge_sum(const int* __restrict__ ei, int E, int ET,
                           const float* __restrict__ a_src, const float* __restrict__ a_dst,
                           const float* __restrict__ mmax, float* __restrict__ ssum, int H) {
    int t = blockIdx.x * blockDim.x + threadIdx.x;
    if (t >= ET * H) return;
    int h = t % H, k = t / H;
    int s, d; edge_sd(ei, E, k, &s, &d);
    float e = a_src[s * H + h] + a_dst[d * H + h];
    e = (e > 0.f) ? e : NEG_SLOPE * e;
    atomicAdd(&ssum[d * H + h], expf(e - mmax[d * H + h]));
}

// scatter-add of alpha-weighted messages; one thread per (edge, head, 16-dim chunk).
// Chunk extents here are always multiples of 4 and 16B-aligned -> float4 gathers.
__global__ void k_edge_agg(const int* __restrict__ ei, int E, int ET,
                           const float* __restrict__ a_src, const float* __restrict__ a_dst,
                           const float* __restrict__ mmax, const float* __restrict__ ssum,
                           const float* __restrict__ feat, float* __restrict__ out,
                           int H, int D, int nchunk) {
    int t = blockIdx.x * blockDim.x + threadIdx.x;
    if (t >= ET * H * nchunk) return;
    int chunk = t % nchunk;
    int r = t / nchunk;
    int h = r % H, k = r / H;
    int s, d; edge_sd(ei, E, k, &s, &d);
    float e = a_src[s * H + h] + a_dst[d * H + h];
    e = (e > 0.f) ? e : NEG_SLOPE * e;
    float alpha = expf(e - mmax[d * H + h]) / (ssum[d * H + h] + 1e-16f);
    int d0 = chunk * 16;
    int d1 = d0 + 16; if (d1 > D) d1 = D;
    const float* fp = feat + (size_t)s * H * D + (size_t)h * D;
    float*       op = out  + (size_t)d * H * D + (size_t)h * D;
    for (int dd = d0; dd + 3 < d1; dd += 4) {
        v4f v = *(const v4f*)(fp + dd);        // 16B-aligned gather, L2-resident
        atomicAdd(&op[dd + 0], v[0] * alpha);
        atomicAdd(&op[dd + 1], v[1] * alpha);
        atomicAdd(&op[dd + 2], v[2] * alpha);
        atomicAdd(&op[dd + 3], v[3] * alpha);
    }
}

// ------------------------------------------------------------ pointwise
__global__ void k_elu_bias(float* __restrict__ p, const float* __restrict__ bias,
                           int n, int cols) {
    int i = blockIdx.x * blockDim.x + threadIdx.x;
    if (i >= n) return;
    float v = p[i] + bias[i % cols];
    p[i] = (v > 0.f) ? v : (expf(v) - 1.f);
}

__global__ void k_logsoftmax(const float* __restrict__ in, const float* __restrict__ bias,
                             float* __restrict__ out, int Nn, int C) {
    int n = blockIdx.x * blockDim.x + threadIdx.x;
    if (n >= Nn) return;
    const float* ip = in + (size_t)n * C;
    float mx = -INFINITY;
    for (int c = 0; c < C; ++c) { float v = ip[c] + bias[c]; if (v > mx) mx = v; }
    float sum = 0.f;
    for (int c = 0; c < C; ++c) sum += expf(ip[c] + bias[c] - mx);
    float lse = mx + logf(sum);
    for (int c = 0; c < C; ++c) out[(size_t)n * C + c] = ip[c] + bias[c] - lse;
}

// ---------------------------------------------------------------- launcher
static inline int cdiv(int a, int b) { return (a + b - 1) / b; }

extern "C" void kernel_launch(void* const* d_in, const int* in_sizes, int n_in,
                              void* d_out, int out_size, void* d_ws, size_t ws_size,
                              hipStream_t stream) {
    const float* x        = (const float*)d_in[0];
    const float* W1       = (const float*)d_in[1];
    const float* att_src1 = (const float*)d_in[2];
    const float* att_dst1 = (const float*)d_in[3];
    const float* bias1    = (const float*)d_in[4];
    const float* W2       = (const float*)d_in[5];
    const float* att_src2 = (const float*)d_in[6];
    const float* att_dst2 = (const float*)d_in[7];
    const float* bias2    = (const float*)d_in[8];
    const int*   ei       = (const int*)d_in[9];

    const int N  = in_sizes[0] / IN_DIM;   // 50000
    const int E  = in_sizes[9] / 2;        // 800000
    const int ET = E + N;                  // + self loops
    const int F1 = HEADS * HID;            // 128

    // workspace layout (floats); total ~66 MB, fits in the 192 MB L2
    float* ws   = (float*)d_ws;
    float* h1   = ws;                          // N*128  (reused as h2: N*40)
    float* out1 = h1   + (size_t)N * F1;       // N*128
    float* as1  = out1 + (size_t)N * F1;       // N*8
    float* ad1  = as1  + (size_t)N * HEADS;    // N*8
    float* m1   = ad1  + (size_t)N * HEADS;    // N*8
    float* s1   = m1   + (size_t)N * HEADS;    // N*8
    float* as2  = s1   + (size_t)N * HEADS;    // N
    float* ad2  = as2  + (size_t)N;            // N
    float* m2   = ad2  + (size_t)N;            // N
    float* s2   = m2   + (size_t)N;            // N
    float* out2 = s2   + (size_t)N;            // N*40
    float* h2   = h1;                          // reuse: h1 dead once agg1 done

    const int BT = 256;

    // --- init accumulators (both layers) ---
    k_fill<<<cdiv(N * HEADS, BT), BT, 0, stream>>>(m1, -INFINITY, N * HEADS);
    k_fill<<<cdiv(N * HEADS, BT), BT, 0, stream>>>(s1, 0.f, N * HEADS);
    k_fill<<<cdiv(N * F1,   BT), BT, 0, stream>>>(out1, 0.f, N * F1);
    k_fill<<<cdiv(N,        BT), BT, 0, stream>>>(m2, -INFINITY, N);
    k_fill<<<cdiv(N,        BT), BT, 0, stream>>>(s2, 0.f, N);
    k_fill<<<cdiv(N * OUT_DIM, BT), BT, 0, stream>>>(out2, 0.f, N * OUT_DIM);

    // --- layer 1: h1 = x @ W1 (WMMA f32), attention, segment softmax, aggregate ---
    {
        int colTiles = cdiv(F1, 16);   // 8 waves/block
        k_gemm_wmma_f32<<<cdiv(N, 16), 32 * colTiles, 0, stream>>>(x, W1, h1, N, IN_DIM, F1);
    }
    k_att<<<cdiv(N * HEADS, BT), BT, 0, stream>>>(h1, att_src1, att_dst1, as1, ad1,
                                                  N, HEADS, HID);
    k_edge_max<<<cdiv(ET * HEADS, BT), BT, 0, stream>>>(ei, E, ET, as1, ad1, m1, HEADS);
    k_edge_sum<<<cdiv(ET * HEADS, BT), BT, 0, stream>>>(ei, E, ET, as1, ad1, m1, s1, HEADS);
    k_edge_agg<<<cdiv(ET * HEADS, BT), BT, 0, stream>>>(ei, E, ET, as1, ad1, m1, s1,
                                                        h1, out1, HEADS, HID, 1);
    // elu(out1 + bias1) in place -> layer-2 input features
    k_elu_bias<<<cdiv(N * F1, BT), BT, 0, stream>>>(out1, bias1, N * F1, F1);

    // --- layer 2: h2 = f1 @ W2 (WMMA f32, 40 cols padded to 3 tiles) ---
    {
        int colTiles = cdiv(OUT_DIM, 16);  // 3 waves/block
        k_gemm_wmma_f32<<<cdiv(N, 16), 32 * colTiles, 0, stream>>>(out1, W2, h2,
                                                                   N, F1, OUT_DIM);
    }
    k_att<<<cdiv(N, BT), BT, 0, stream>>>(h2, att_src2, att_dst2, as2, ad2, N, 1, OUT_DIM);
    k_edge_max<<<cdiv(ET, BT), BT, 0, stream>>>(ei, E, ET, as2, ad2, m2, 1);
    k_edge_sum<<<cdiv(ET, BT), BT, 0, stream>>>(ei, E, ET, as2, ad2, m2, s2, 1);
    {
        int nchunk = cdiv(OUT_DIM, 16);    // 3 chunks: 16,16,8
        k_edge_agg<<<cdiv(ET * nchunk, BT), BT, 0, stream>>>(ei, E, ET, as2, ad2, m2, s2,
                                                             h2, out2, 1, OUT_DIM, nchunk);
    }

    // --- heads=1 mean is identity; add bias2 and log_softmax ---
    k_logsoftmax<<<cdiv(N, BT), BT, 0, stream>>>(out2, bias2, (float*)d_out, N, OUT_DIM);
}